// ButterflyFactorNewMlp_47304769798187
// MI455X (gfx1250) — compile-verified
//
#include <hip/hip_runtime.h>
#include <math.h>

// ---------------------------------------------------------------------------
// Problem constants
// ---------------------------------------------------------------------------
#define IN_F   729
#define HID    2916
#define OUT_F  729
#define MROWS  49152     // B * D = 64 * 768
#define ALD    768       // padded attn leading dim (and padded row count)

// ---------------------------------------------------------------------------
// Tiling
// ---------------------------------------------------------------------------
#define BK    16
#define LDA   17         // As row stride (floats): 17*m mod 64 distinct -> no conflicts
// kernel 1 (small GEMM)
#define BN1   128
#define LDB1  132
// kernel 2 (main GEMM)
#define BN2   256
#define LDB2  260        // multiple of 4 -> 16B-aligned rows for b128 staging
#define ASZ   (64 * LDA)   // one A buffer (floats)
#define BSZ   (BK * LDB2)  // one B buffer (floats)

typedef __attribute__((ext_vector_type(2))) float v2f;
typedef __attribute__((ext_vector_type(8))) float v8f;
typedef int vs_i32x4 __attribute__((vector_size(16)));   // matches builtin param type

// ---------------------------------------------------------------------------
// gfx1250 async global->LDS path (ASYNCcnt), with sync fallback
// ---------------------------------------------------------------------------
#if defined(__AMDGCN__) && \
    __has_builtin(__builtin_amdgcn_global_load_async_to_lds_b32) && \
    __has_builtin(__builtin_amdgcn_global_load_async_to_lds_b128) && \
    __has_builtin(__builtin_amdgcn_s_wait_asynccnt)
#define HAS_ASYNC 1
#else
#define HAS_ASYNC 0
#endif

#if HAS_ASYNC
// global (AS1) typed pointers; HIP prints AS1 as __device__
#define ASG32(p)   ((__attribute__((address_space(1))) int*)(void*)(p))
#define ASG128(p)  ((__attribute__((address_space(1))) vs_i32x4*)(void*)(p))
// LDS (AS3) typed pointers
#define ASL32(p)   ((__attribute__((address_space(3))) int*)(void*)(p))
#define ASL128(p)  ((__attribute__((address_space(3))) vs_i32x4*)(void*)(p))
#define WAIT_ASYNC(n) __builtin_amdgcn_s_wait_asynccnt(n)
#else
#define WAIT_ASYNC(n)
#endif

__device__ __forceinline__ float gelu_exact(float v) {
    return 0.5f * v * (1.0f + erff(v * 0.70710678118654752440f));
}

// ---------------------------------------------------------------------------
// Kernel 1: attnT = w2(729x2916) * w1(2916x729); store transposed + masked
// into zero-padded attn[768][768]:  attn[s*ALD+t] = attnT[t,s]*mask[s,t]
// (padding rows/cols are written as zeros so kernel 2 needs no B guards)
// ---------------------------------------------------------------------------
__global__ __launch_bounds__(256)
void bfnm_attn_kernel(const float* __restrict__ w1,
                      const float* __restrict__ w2,
                      const float* __restrict__ mask,
                      float* __restrict__ attn) {
    __shared__ float As[64 * LDA];
    __shared__ float Bs[BK * LDB1];

    const int tid  = threadIdx.x;
    const int lane = tid & 31;
    const int wave = tid >> 5;
    const int wm   = (wave >> 2) * 32;
    const int wn   = (wave & 3) * 32;
    const int l16  = lane & 15;
    const int lh   = lane >> 4;

    const int rowBlock = blockIdx.y * 64;    // t
    const int colBlock = blockIdx.x * BN1;   // s

    v8f acc[2][2] = {};

    const int ksteps = (HID + BK - 1) / BK;  // 183
    for (int kb0 = 0; kb0 < ksteps; ++kb0) {
        const int kb = kb0 * BK;
        // A[t,d] = w2[t*HID+d] (64x16), branchless clamp+select guards
        #pragma unroll
        for (int i = 0; i < 4; ++i) {
            int idx = tid + i * 256;
            int rr = idx >> 4, kk = idx & 15;
            int t = rowBlock + rr, d = kb + kk;
            int tc = t < OUT_F ? t : OUT_F - 1;
            int dc = d < HID  ? d : HID  - 1;
            float v = w2[tc * HID + dc];
            As[rr * LDA + kk] = (t < OUT_F && d < HID) ? v : 0.0f;
        }
        // B[d,s] = w1[d*IN_F+s] (16x128)
        #pragma unroll
        for (int i = 0; i < 8; ++i) {
            int idx = tid + i * 256;
            int kr = idx >> 7, nn = idx & 127;
            int d = kb + kr, s = colBlock + nn;
            int dc = d < HID  ? d : HID  - 1;
            int sc = s < IN_F ? s : IN_F - 1;
            float v = w1[dc * IN_F + sc];
            Bs[kr * LDB1 + nn] = (d < HID && s < IN_F) ? v : 0.0f;
        }
        __syncthreads();
        #pragma unroll
        for (int ks = 0; ks < 4; ++ks) {
            const int k = ks * 4;
            v2f a[2], b[2];
            #pragma unroll
            for (int mt = 0; mt < 2; ++mt) {
                int r = wm + mt * 16 + l16;
                a[mt].x = As[r * LDA + k + 2 * lh + 0];
                a[mt].y = As[r * LDA + k + 2 * lh + 1];
            }
            #pragma unroll
            for (int nt = 0; nt < 2; ++nt) {
                int c = wn + nt * 16 + l16;
                b[nt].x = Bs[(k + 2 * lh + 0) * LDB1 + c];
                b[nt].y = Bs[(k + 2 * lh + 1) * LDB1 + c];
            }
            #pragma unroll
            for (int mt = 0; mt < 2; ++mt)
                #pragma unroll
                for (int nt = 0; nt < 2; ++nt)
                    acc[mt][nt] = __builtin_amdgcn_wmma_f32_16x16x4_f32(
                        false, a[mt], false, b[nt],
                        (short)0, acc[mt][nt], false, false);
        }
        __syncthreads();
    }

    // Epilogue: write full padded tile (zeros outside 729x729)
    #pragma unroll
    for (int mt = 0; mt < 2; ++mt) {
        #pragma unroll
        for (int nt = 0; nt < 2; ++nt) {
            #pragma unroll
            for (int v = 0; v < 8; ++v) {
                int t = rowBlock + wm + mt * 16 + v + 8 * lh;
                int s = colBlock + wn + nt * 16 + l16;
                int tc = t < OUT_F ? t : OUT_F - 1;
                int sc = s < IN_F ? s : IN_F - 1;
                float m   = mask[sc * OUT_F + tc];
                float val = (t < OUT_F && s < IN_F) ? acc[mt][nt][v] * m : 0.0f;
                attn[s * ALD + t] = val;
            }
        }
    }
}

// ---------------------------------------------------------------------------
// Kernel 2: out[r,t] = gelu( sum_s x[r,s]*attn[s,t] + b2[t] )
// 64x256 block tile, 8 waves of 32x64, double-buffered async LDS staging.
// ---------------------------------------------------------------------------
__global__ __launch_bounds__(256)
void bfnm_mlp_kernel(const float* __restrict__ x,
                     const float* __restrict__ attn,
                     const float* __restrict__ b2,
                     float* __restrict__ out) {
    __shared__ float lds[2 * ASZ + 2 * BSZ];

    const int tid  = threadIdx.x;
    const int lane = tid & 31;
    const int wave = tid >> 5;
    const int wm   = (wave >> 2) * 32;   // 0 / 32
    const int wn   = (wave & 3) * 64;    // 0..192
    const int l16  = lane & 15;
    const int lh   = lane >> 4;

    const int rowBlock = blockIdx.y * 64;    // r, always fully in range
    const int colBlock = blockIdx.x * BN2;   // t

    v8f acc[2][4] = {};

    // B tile staging (16 x 256): attn is padded -> no guards, 16B aligned
    auto stage_B = [&](int kb, float* B) {
#if HAS_ASYNC
        #pragma unroll
        for (int i = 0; i < 4; ++i) {
            int idx = tid + i * 256;
            int kr = idx >> 6, c4 = (idx & 63) << 2;
            __builtin_amdgcn_global_load_async_to_lds_b128(
                ASG128(attn + (kb + kr) * ALD + colBlock + c4),
                ASL128(B + kr * LDB2 + c4), 0, 0);
        }
#else
        #pragma unroll
        for (int i = 0; i < 4; ++i) {
            int idx = tid + i * 256;
            int kr = idx >> 6, c4 = (idx & 63) << 2;
            const float4 v = *(const float4*)(attn + (kb + kr) * ALD + colBlock + c4);
            *(float4*)(B + kr * LDB2 + c4) = v;
        }
#endif
    };
#if HAS_ASYNC
    // A tile staging (64 x 16) from x, unguarded full steps: async b32
    auto stage_A_async = [&](int kb, float* A) {
        #pragma unroll
        for (int i = 0; i < 4; ++i) {
            int idx = tid + i * 256;
            int rr = idx >> 4, kk = idx & 15;
            __builtin_amdgcn_global_load_async_to_lds_b32(
                ASG32(x + (rowBlock + rr) * IN_F + kb + kk),
                ASL32(A + rr * LDA + kk), 0, 0);
        }
    };
#endif
    // A tile staging, sync; guard=true only for the single K tail step
    auto stage_A_sync = [&](int kb, float* A, bool guard) {
        #pragma unroll
        for (int i = 0; i < 4; ++i) {
            int idx = tid + i * 256;
            int rr = idx >> 4, kk = idx & 15;
            int s = kb + kk;
            if (!guard) {
                A[rr * LDA + kk] = x[(rowBlock + rr) * IN_F + s];
            } else {
                int sc = s < IN_F ? s : IN_F - 1;
                float v = x[(rowBlock + rr) * IN_F + sc];
                A[rr * LDA + kk] = (s < IN_F) ? v : 0.0f;
            }
        }
    };

    constexpr int KSTEPS = 46;   // ceil(729/16); LDS zero-padding covers tail
    constexpr int KFULL  = 45;   // steps with no x-guard needed (kb+15 <= 728)

    // Prologue: stage step 0 into buffer 0
#if HAS_ASYNC
    stage_A_async(0, lds);
#else
    stage_A_sync(0, lds, false);
#endif
    stage_B(0, lds + 2 * ASZ);

    for (int i = 0; i < KSTEPS; ++i) {
        const int ib = i & 1;
        const int nb = ib ^ 1;
        float* Ac = lds + ib * ASZ;
        float* An = lds + nb * ASZ;
        float* Bc = lds + 2 * ASZ + ib * BSZ;
        float* Bn = lds + 2 * ASZ + nb * BSZ;
        const int next = i + 1;

        if (next < KSTEPS) {
            stage_B(next * BK, Bn);
            if (next < KFULL) {
#if HAS_ASYNC
                stage_A_async(next * BK, An);
                WAIT_ASYNC(8);   // >8 outstanding -> step i's 8 asyncs done (in-order)
#else
                stage_A_sync(next * BK, An, false);
#endif
            } else {
                stage_A_sync(next * BK, An, true);   // guarded tail, ds stores
                WAIT_ASYNC(4);   // only next step's 4 B-asyncs may remain
            }
        } else {
            WAIT_ASYNC(0);
        }
        __syncthreads();

        #pragma unroll
        for (int ks = 0; ks < 4; ++ks) {
            const int k = ks * 4;
            v2f a[2], b[4];
            #pragma unroll
            for (int mt = 0; mt < 2; ++mt) {
                int r = wm + mt * 16 + l16;
                a[mt].x = Ac[r * LDA + k + 2 * lh + 0];
                a[mt].y = Ac[r * LDA + k + 2 * lh + 1];
            }
            #pragma unroll
            for (int nt = 0; nt < 4; ++nt) {
                int c = wn + nt * 16 + l16;
                b[nt].x = Bc[(k + 2 * lh + 0) * LDB2 + c];
                b[nt].y = Bc[(k + 2 * lh + 1) * LDB2 + c];
            }
            #pragma unroll
            for (int mt = 0; mt < 2; ++mt)
                #pragma unroll
                for (int nt = 0; nt < 4; ++nt)
                    acc[mt][nt] = __builtin_amdgcn_wmma_f32_16x16x4_f32(
                        false, a[mt], false, b[nt],
                        (short)0, acc[mt][nt], false, false);
        }
        __syncthreads();
    }

    // Epilogue: bias + exact GELU
    #pragma unroll
    for (int mt = 0; mt < 2; ++mt) {
        #pragma unroll
        for (int nt = 0; nt < 4; ++nt) {
            #pragma unroll
            for (int v = 0; v < 8; ++v) {
                int r = rowBlock + wm + mt * 16 + v + 8 * lh;
                int t = colBlock + wn + nt * 16 + l16;
                if (t < OUT_F) {
                    float val = acc[mt][nt][v] + b2[t];
                    out[(size_t)r * OUT_F + t] = gelu_exact(val);
                }
            }
        }
    }
}

// ---------------------------------------------------------------------------
// Host-side launcher
// ---------------------------------------------------------------------------
extern "C" void kernel_launch(void* const* d_in, const int* in_sizes, int n_in,
                              void* d_out, int out_size, void* d_ws, size_t ws_size,
                              hipStream_t stream) {
    const float* x    = (const float*)d_in[0];   // [64, 768, 729]
    const float* w1   = (const float*)d_in[1];   // [2916, 729]
    const float* w2   = (const float*)d_in[2];   // [729, 2916]
    const float* b2   = (const float*)d_in[3];   // [729]
    const float* mask = (const float*)d_in[4];   // [729, 729]
    float* out  = (float*)d_out;                 // [64, 768, 729]
    float* attn = (float*)d_ws;                  // padded [768, 768] scratch

    (void)in_sizes; (void)n_in; (void)out_size; (void)ws_size;

    // Kernel 1: masked attn (729x729x2916 GEMM) -> zero-padded 768x768
    dim3 g1(ALD / BN1, ALD / 64);   // (6, 12) covers full padded tile
    bfnm_attn_kernel<<<g1, dim3(256), 0, stream>>>(w1, w2, mask, attn);

    // Kernel 2: main GEMM + bias + exact GELU (49152 x 729 x 729)
    dim3 g2(ALD / BN2, MROWS / 64); // (3, 768)
    bfnm_mlp_kernel<<<g2, dim3(256), 0, stream>>>(x, attn, b2, out);
}